// AreaGNN_15247133900959
// MI455X (gfx1250) — compile-verified
//
#include <hip/hip_runtime.h>
#include <hip/hip_bf16.h>

#define NN 50000
#define EE 800000
#define HID 128
#define GG 64
#define GF 32
#define BN_EPS 1e-5f

typedef __attribute__((ext_vector_type(2))) float v2f;
typedef __attribute__((ext_vector_type(8))) float v8f;

// ---------------- generic fill ----------------
__global__ void fill_kernel(float* __restrict__ p, float v, int n) {
    int i = blockIdx.x * blockDim.x + threadIdx.x;
    if (i < n) p[i] = v;
}

// ------------- edge scatter: mean-aggregate numerator + degree -------------
// one thread per (edge, 4-feature chunk); a wave32 covers one full 512B row
// of h[src] (perfectly coalesced) and hits 32 consecutive atomic targets.
__global__ void scatter_kernel(const float* __restrict__ h,
                               const int* __restrict__ ei,
                               float* __restrict__ agg,
                               float* __restrict__ cnt) {
    int tid = blockIdx.x * blockDim.x + threadIdx.x;
    if (tid >= EE * 32) return;
    int e = tid >> 5;
    int c = (tid & 31) << 2;
    int s = ei[e];          // src row of edge_index
    int d = ei[EE + e];     // dst row
    const float4 v = *(const float4*)(h + (size_t)s * HID + c);
    float* a = agg + (size_t)d * HID + c;
    atomicAdd(a + 0, v.x);
    atomicAdd(a + 1, v.y);
    atomicAdd(a + 2, v.z);
    atomicAdd(a + 3, v.w);
    if (c == 0) atomicAdd(&cnt[d], 1.0f);
}

__global__ void invcnt_kernel(const float* __restrict__ cnt, float* __restrict__ inv) {
    int i = blockIdx.x * blockDim.x + threadIdx.x;
    if (i < NN) inv[i] = 1.0f / fmaxf(cnt[i], 1.0f);
}

// ------------- fused SAGE GEMM: out = (agg*inv) @ Wl + h @ Wr + bl -------------
// wave32; one wave owns a 16-row block and all 8 column tiles (128 cols).
// Both weight matrices (2 x 64KB) are staged once per block into dynamic LDS
// (128KB, within CDNA5's 320KB/WGP), so the hot loop's B-fragment loads ride
// the DS pipe while A-fragment gathers use VMEM and accumulate on the WMMA pipe.
// A fragment (16x4 f32, 2 VGPRs): lanes 0-15 -> M=lane, K={0,1}; lanes 16-31 -> K={2,3}.
// B fragment (4x16 f32, 2 VGPRs): VGPR r: lanes 0-15 K=r, lanes 16-31 K=r+2, N=lane&15.
// C/D (16x16 f32, 8 VGPRs): VGPR r: lanes 0-15 -> (M=r, N=lane); lanes 16-31 -> (M=r+8).
// In-place safe: each wave reads only its own 16 rows of agg, stores only at the end.
__global__ __launch_bounds__(128)
void sage_gemm_wmma_kernel(const float* __restrict__ agg,
                           const float* __restrict__ inv,
                           const float* __restrict__ h,
                           const float* __restrict__ Wl,
                           const float* __restrict__ Wr,
                           const float* __restrict__ bl,
                           float* __restrict__ out) {
    extern __shared__ float smem[];     // [0..16383]=Wl, [16384..32767]=Wr

    // stage weights: 128 threads x 256 floats each, as float4
    {
        float4* sw = (float4*)smem;
        const float4* gl = (const float4*)Wl;
        const float4* gr = (const float4*)Wr;
        for (int i = threadIdx.x; i < (HID * HID) / 4; i += 128) {
            sw[i] = gl[i];
            sw[(HID * HID) / 4 + i] = gr[i];
        }
    }
    __syncthreads();

    const int lane = threadIdx.x & 31;
    const int wave = threadIdx.x >> 5;
    const int rowblk = blockIdx.x * 4 + wave;
    const int row0 = rowblk * 16;
    if (row0 >= NN) return;             // wave-uniform: EXEC stays all-ones

    const float* sWl = smem;
    const float* sWr = smem + HID * HID;

    const int half = lane >> 4;         // which K-pair this lane holds
    const int l16  = lane & 15;
    const int mrow = row0 + l16;
    const float sc = inv[mrow];         // fold mean-divide into A load

    v8f acc[8];
#pragma unroll
    for (int t = 0; t < 8; ++t) {
        const float b = bl[t * 16 + l16];   // bias broadcast along rows
#pragma unroll
        for (int r = 0; r < 8; ++r) acc[t][r] = b;
    }

    const float* arow_agg = agg + (size_t)mrow * HID;
    const float* arow_h   = h   + (size_t)mrow * HID;

    for (int k0 = 0; k0 < HID; k0 += 4) {
        const int ka = k0 + 2 * half;
        v2f a_agg, a_h;
        a_agg.x = arow_agg[ka]     * sc;
        a_agg.y = arow_agg[ka + 1] * sc;
        a_h.x   = arow_h[ka];
        a_h.y   = arow_h[ka + 1];
#pragma unroll
        for (int t = 0; t < 8; ++t) {
            const int col = t * 16 + l16;
            v2f b_l, b_r;
            b_l.x = sWl[ka * HID + col];
            b_l.y = sWl[(ka + 1) * HID + col];
            b_r.x = sWr[ka * HID + col];
            b_r.y = sWr[(ka + 1) * HID + col];
            acc[t] = __builtin_amdgcn_wmma_f32_16x16x4_f32(
                false, a_agg, false, b_l, (short)0, acc[t], false, false);
            acc[t] = __builtin_amdgcn_wmma_f32_16x16x4_f32(
                false, a_h, false, b_r, (short)0, acc[t], false, false);
        }
    }

#pragma unroll
    for (int t = 0; t < 8; ++t) {
        const int col = t * 16 + l16;
#pragma unroll
        for (int r = 0; r < 8; ++r) {
            out[(size_t)(row0 + r + 8 * half) * HID + col] = acc[t][r];
        }
    }
}

// ------------- BatchNorm over nodes (two-pass) + ReLU, in place -------------
__global__ void bn_stats_kernel(const float* __restrict__ y, float* __restrict__ stats) {
    const int f = threadIdx.x;              // 128 features
    float s = 0.0f, s2 = 0.0f;
    for (int r = blockIdx.x; r < NN; r += gridDim.x) {
        const float v = y[(size_t)r * HID + f];
        s += v; s2 += v * v;
    }
    atomicAdd(&stats[f], s);
    atomicAdd(&stats[HID + f], s2);
}

__global__ void bn_apply_kernel(float* __restrict__ y, const float* __restrict__ stats,
                                const float* __restrict__ gamma,
                                const float* __restrict__ beta) {
    int tid = blockIdx.x * blockDim.x + threadIdx.x;
    if (tid >= NN * HID) return;
    const int f = tid & (HID - 1);
    const float invN = 1.0f / (float)NN;
    const float mu  = stats[f] * invN;
    const float var = stats[HID + f] * invN - mu * mu;
    const float v = (y[tid] - mu) * rsqrtf(var + BN_EPS);
    y[tid] = fmaxf(gamma[f] * v + beta[f], 0.0f);
}

// ------------- global pooling: per-graph sum/max/count -------------
__global__ void pool_kernel(const float* __restrict__ h, const int* __restrict__ batch,
                            float* __restrict__ gsum, unsigned* __restrict__ gmax_u,
                            float* __restrict__ gcnt) {
    int tid = blockIdx.x * blockDim.x + threadIdx.x;
    if (tid >= NN * HID) return;
    const int node = tid >> 7;
    const int f = tid & (HID - 1);
    const int g = batch[node];
    const float v = h[(size_t)node * HID + f];
    atomicAdd(&gsum[g * HID + f], v);
    atomicMax(&gmax_u[g * HID + f], __float_as_uint(v));  // v >= 0 post-ReLU
    if (f == 0) atomicAdd(&gcnt[g], 1.0f);
}

// ------------- head MLP (tiny: G=64) -------------
__global__ void head1_kernel(const float* __restrict__ gsum, const float* __restrict__ gmax,
                             const float* __restrict__ gcnt, const float* __restrict__ gfeat,
                             const float* __restrict__ W1, const float* __restrict__ b1,
                             float* __restrict__ o1) {
    const int g = blockIdx.x, j = threadIdx.x;  // 64 x 128
    const float invc = 1.0f / fmaxf(gcnt[g], 1.0f);
    float s = b1[j];
    for (int k = 0; k < HID; ++k) s += gsum[g * HID + k] * invc * W1[k * HID + j];
    for (int k = 0; k < HID; ++k) s += gmax[g * HID + k] * W1[(HID + k) * HID + j];
    for (int k = 0; k < GF; ++k)  s += gfeat[g * GF + k] * W1[(2 * HID + k) * HID + j];
    o1[g * HID + j] = fmaxf(s, 0.0f);
}

__global__ void head2_kernel(const float* __restrict__ o1, const float* __restrict__ W2,
                             const float* __restrict__ b2, float* __restrict__ o2) {
    const int g = blockIdx.x, j = threadIdx.x;  // 64 x 64
    float s = b2[j];
    for (int k = 0; k < HID; ++k) s += o1[g * HID + k] * W2[k * 64 + j];
    o2[g * 64 + j] = fmaxf(s, 0.0f);
}

__global__ void head3_kernel(const float* __restrict__ o2, const float* __restrict__ W3,
                             const float* __restrict__ b3, float* __restrict__ out) {
    const int g = threadIdx.x;  // 64
    if (g >= GG) return;
    float s = b3[0];
    for (int k = 0; k < 64; ++k) s += o2[g * 64 + k] * W3[k];
    out[g] = s;
}

extern "C" void kernel_launch(void* const* d_in, const int* in_sizes, int n_in,
                              void* d_out, int out_size, void* d_ws, size_t ws_size,
                              hipStream_t stream) {
    const float* x     = (const float*)d_in[0];
    const int*   ei    = (const int*)d_in[1];
    const int*   batch = (const int*)d_in[2];
    const float* gfeat = (const float*)d_in[3];
    const float* Wl[3] = {(const float*)d_in[4],  (const float*)d_in[9],  (const float*)d_in[14]};
    const float* blv[3]= {(const float*)d_in[5],  (const float*)d_in[10], (const float*)d_in[15]};
    const float* Wr[3] = {(const float*)d_in[6],  (const float*)d_in[11], (const float*)d_in[16]};
    const float* gam[3]= {(const float*)d_in[7],  (const float*)d_in[12], (const float*)d_in[17]};
    const float* bet[3]= {(const float*)d_in[8],  (const float*)d_in[13], (const float*)d_in[18]};
    const float* W1 = (const float*)d_in[19];
    const float* b1 = (const float*)d_in[20];
    const float* W2 = (const float*)d_in[21];
    const float* b2 = (const float*)d_in[22];
    const float* W3 = (const float*)d_in[23];
    const float* b3 = (const float*)d_in[24];

    // workspace layout (floats)
    float* ws    = (float*)d_ws;
    float* P     = ws;                    // N*HID
    float* Q     = P + (size_t)NN * HID;  // N*HID
    float* cnt   = Q + (size_t)NN * HID;  // N
    float* inv   = cnt + NN;              // N
    float* stats = inv + NN;              // 2*HID
    float* gsum  = stats + 2 * HID;       // G*HID
    float* gmaxf = gsum + GG * HID;       // G*HID (uint bits)
    float* gcnt  = gmaxf + GG * HID;      // G
    float* o1    = gcnt + GG;             // G*HID
    float* o2    = o1 + GG * HID;         // G*64

    const int TPB = 256;
    const int nh = NN * HID;
    const int gemm_blocks = ((NN / 16) + 3) / 4;          // 4 waves/block, 16 rows/wave
    const size_t lds_bytes = 2 * HID * HID * sizeof(float); // 128KB (<= 320KB/WGP)

    auto layer = [&](const float* hin, float* aggbuf, int li) {
        fill_kernel<<<(nh + TPB - 1) / TPB, TPB, 0, stream>>>(aggbuf, 0.0f, nh);
        fill_kernel<<<(NN + TPB - 1) / TPB, TPB, 0, stream>>>(cnt, 0.0f, NN);
        scatter_kernel<<<(EE * 32 + TPB - 1) / TPB, TPB, 0, stream>>>(hin, ei, aggbuf, cnt);
        invcnt_kernel<<<(NN + TPB - 1) / TPB, TPB, 0, stream>>>(cnt, inv);
        sage_gemm_wmma_kernel<<<gemm_blocks, 128, lds_bytes, stream>>>(
            aggbuf, inv, hin, Wl[li], Wr[li], blv[li], aggbuf);
        fill_kernel<<<1, TPB, 0, stream>>>(stats, 0.0f, 2 * HID);
        bn_stats_kernel<<<512, HID, 0, stream>>>(aggbuf, stats);
        bn_apply_kernel<<<(nh + TPB - 1) / TPB, TPB, 0, stream>>>(
            aggbuf, stats, gam[li], bet[li]);
    };

    layer(x, P, 0);   // h1 = P
    layer(P, Q, 1);   // h2 = Q
    layer(Q, P, 2);   // h3 = P

    fill_kernel<<<(GG * HID + TPB - 1) / TPB, TPB, 0, stream>>>(gsum, 0.0f, GG * HID);
    fill_kernel<<<(GG * HID + TPB - 1) / TPB, TPB, 0, stream>>>(gmaxf, 0.0f, GG * HID);
    fill_kernel<<<1, GG, 0, stream>>>(gcnt, 0.0f, GG);
    pool_kernel<<<(nh + TPB - 1) / TPB, TPB, 0, stream>>>(P, batch, gsum, (unsigned*)gmaxf, gcnt);

    head1_kernel<<<GG, HID, 0, stream>>>(gsum, gmaxf, gcnt, gfeat, W1, b1, o1);
    head2_kernel<<<GG, 64, 0, stream>>>(o1, W2, b2, o2);
    head3_kernel<<<1, 64, 0, stream>>>(o2, W3, b3, (float*)d_out);
}